// ScaledDotProductAttention_45921790329377
// MI455X (gfx1250) — compile-verified
//
#include <hip/hip_runtime.h>
#include <hip/hip_bf16.h>

typedef __attribute__((ext_vector_type(2))) float v2f;
typedef __attribute__((ext_vector_type(4))) float v4f;
typedef __attribute__((ext_vector_type(8))) float v8f;

#define B_  32
#define L_  2048
#define D_  64
#define QT  16          // q rows per block tile
#define NW  4           // waves per block
#define NT  128         // threads per block

__global__ __launch_bounds__(NT, 1)
void sdpa_f32_wmma_kernel(const float* __restrict__ Q,
                          const float* __restrict__ K,
                          const float* __restrict__ V,
                          const int*   __restrict__ diag_mask,
                          const unsigned char* __restrict__ mask,
                          float* __restrict__ out,    // [B, L, D]
                          float* __restrict__ attn)   // [B, L, L]
{
    // 16 x 2048 fp32 score/prob tile: 128 KB (CDNA5 WGP has 320 KB LDS)
    __shared__ float sc[QT * L_];
    __shared__ float part[NW][QT * D_];   // 16 KB PV partials
    __shared__ float red[QT][8];
    __shared__ float rowmax[QT];
    __shared__ float rowinv[QT];

    const int tid  = threadIdx.x;
    const int wave = tid >> 5;
    const int lane = tid & 31;
    const int half = lane >> 4;          // 0: lanes 0-15, 1: lanes 16-31
    const int l15  = lane & 15;

    const int bid = blockIdx.x;
    const int b   = bid >> 7;            // / (L_/QT)
    const int q0  = (bid & 127) << 4;

    const float* Qb = Q + (size_t)b * L_ * D_;
    const float* Kb = K + (size_t)b * L_ * D_;
    const float* Vb = V + (size_t)b * L_ * D_;

    // ---------------- Load Q A-fragments once (A: 16x4 f32 per step) -------
    // ISA layout: lanes 0-15 -> M=0..15 with K = {4t+0, 4t+1};
    //             lanes 16-31 -> same M with K = {4t+2, 4t+3}.
    v2f aq[16];
    {
        const float* qrow = Qb + (size_t)(q0 + l15) * D_;
        #pragma unroll
        for (int t = 0; t < 16; ++t) {
            const int d0 = t * 4 + half * 2;
            aq[t].x = qrow[d0];
            aq[t].y = qrow[d0 + 1];
        }
    }

    // ---------------- Phase 1: scores = (Q K^T)/8 + masks -> LDS -----------
    // Each wave owns 32 contiguous 16-wide key tiles.
    for (int kt = wave * 32; kt < wave * 32 + 32; ++kt) {
        const int kbase = kt * 16;

        // prefetch next key tile (speculative; emits global_prefetch_b8)
        __builtin_prefetch(Kb + (size_t)(kbase + 16 + l15) * D_, 0, 0);

        v8f c = {};
        const float* krow = Kb + (size_t)(kbase + l15) * D_;
        #pragma unroll
        for (int t = 0; t < 16; ++t) {
            const int d0 = t * 4 + half * 2;
            // B: 4x16 f32; lane l -> column N=l&15, rows K = {2*half, 2*half+1}
            v2f bf;
            bf.x = krow[d0];
            bf.y = krow[d0 + 1];
            c = __builtin_amdgcn_wmma_f32_16x16x4_f32(
                    false, aq[t], false, bf, (short)0, c, false, false);
        }

        // C layout: VGPR r -> M = r + 8*half, N = l15
        // mask / diag_mask are streamed read-once (640 MB): use non-temporal
        // loads so they don't evict the L2-resident Q/K/V working set.
        #pragma unroll
        for (int r = 0; r < 8; ++r) {
            const int m  = r + half * 8;
            const int kk = kbase + l15;
            float s = c[r] * 0.125f;                       // / temperature (8)
            const size_t mi = (size_t)b * L_ * L_ + (size_t)(q0 + m) * L_ + kk;
            const unsigned char mk = __builtin_nontemporal_load(&mask[mi]);
            const int           dg = __builtin_nontemporal_load(&diag_mask[mi]);
            if (mk)      s = -__builtin_inff();            // masked_fill(mask, -inf)
            if (dg == 0) s = -1e32f;                       // masked_fill(diag==0, -1e32)
            sc[m * L_ + kk] = s;
        }
    }
    __syncthreads();

    // ---------------- Phase 2: deterministic softmax stats ------------------
    {   // row max: 8 threads per row, contiguous 256-wide chunks
        const int r = tid >> 3, cc = tid & 7;
        const float* rowp = &sc[r * L_ + cc * 256];
        float mx = -__builtin_inff();
        for (int i = 0; i < 256; ++i) mx = fmaxf(mx, rowp[i]);
        red[r][cc] = mx;
    }
    __syncthreads();
    if (tid < QT) {
        float mx = red[tid][0];
        #pragma unroll
        for (int i = 1; i < 8; ++i) mx = fmaxf(mx, red[tid][i]);
        rowmax[tid] = mx;
    }
    __syncthreads();
    {   // exp + row sum (writes exp back to LDS)
        const int r = tid >> 3, cc = tid & 7;
        const float mx = rowmax[r];
        float* rowp = &sc[r * L_ + cc * 256];
        float sum = 0.f;
        for (int i = 0; i < 256; ++i) {
            const float e = __expf(rowp[i] - mx);
            rowp[i] = e;
            sum += e;
        }
        red[r][cc] = sum;
    }
    __syncthreads();
    if (tid < QT) {
        float s = red[tid][0];
        #pragma unroll
        for (int i = 1; i < 8; ++i) s += red[tid][i];
        rowinv[tid] = 1.0f / s;
    }
    __syncthreads();

    // ---------------- Phase 3a: normalize P in LDS (coalesced float4) ------
    {
        v4f* sc4 = (v4f*)sc;
        for (int i = tid; i < QT * L_ / 4; i += NT) {
            const float inv = rowinv[i >> 9];   // 512 float4 per row
            v4f p = sc4[i];
            p.x *= inv; p.y *= inv; p.z *= inv; p.w *= inv;
            sc4[i] = p;                          // normalized P for PV + attn
        }
    }
    __syncthreads();   // make normalized LDS visible to async engine + waves

    // ---------------- Phase 3b: async DMA attn slab LDS -> global -----------
    // This block's attn region is one contiguous 16*2048 fp32 slab that
    // exactly mirrors sc[]. Issue GLOBAL_STORE_ASYNC_FROM_LDS_B128 per lane
    // (ASYNCcnt path) and let the DMA overlap the PV WMMA phase below.
    {
        float* attn_base = attn + ((size_t)b * L_ + q0) * L_;
        for (int i = tid; i < QT * L_ / 4; i += NT) {
            unsigned long long ga = (unsigned long long)(attn_base + (size_t)i * 4);
            unsigned           la = (unsigned)(unsigned long long)&sc[(size_t)i * 4];
            asm volatile("global_store_async_from_lds_b128 %0, %1, off"
                         :: "v"(ga), "v"(la) : "memory");
        }
    }

    // ---------------- Phase 4: out = P @ V (overlapped with attn DMA) ------
    {
        v8f cpv[4] = {{}, {}, {}, {}};           // 4 d-tiles of 16
        const int kstart = wave * (L_ / NW);     // 512 keys per wave
        for (int kk = kstart; kk < kstart + L_ / NW; kk += 4) {
            // A from LDS P: M = l15, K = kk + 2*half + {0,1}
            v2f ap;
            ap.x = sc[l15 * L_ + kk + half * 2];
            ap.y = sc[l15 * L_ + kk + half * 2 + 1];
            const float* v0 = Vb + (size_t)(kk + half * 2) * D_ + l15;
            #pragma unroll
            for (int dt = 0; dt < 4; ++dt) {
                // B: rows K = {2*half, 2*half+1}, column N = dt*16 + l15
                v2f bf;
                bf.x = v0[dt * 16];
                bf.y = v0[dt * 16 + D_];
                cpv[dt] = __builtin_amdgcn_wmma_f32_16x16x4_f32(
                              false, ap, false, bf, (short)0, cpv[dt], false, false);
            }
        }
        #pragma unroll
        for (int dt = 0; dt < 4; ++dt) {
            #pragma unroll
            for (int r = 0; r < 8; ++r) {
                const int m = r + half * 8;
                part[wave][m * D_ + dt * 16 + l15] = cpv[dt][r];
            }
        }
    }
    __syncthreads();

    {   // cross-wave reduce + coalesced contiguous output store
        float* ob = out + ((size_t)b * L_ + q0) * D_;
        for (int e = tid; e < QT * D_; e += NT)
            ob[e] = part[0][e] + part[1][e] + part[2][e] + part[3][e];
    }

    // Drain the async attn DMA before wave termination.
    asm volatile("s_wait_asynccnt 0x0" ::: "memory");
}

extern "C" void kernel_launch(void* const* d_in, const int* in_sizes, int n_in,
                              void* d_out, int out_size, void* d_ws, size_t ws_size,
                              hipStream_t stream) {
    (void)in_sizes; (void)n_in; (void)out_size; (void)d_ws; (void)ws_size;
    const float*         q    = (const float*)d_in[0];
    const float*         k    = (const float*)d_in[1];
    const float*         v    = (const float*)d_in[2];
    const int*           diag = (const int*)d_in[3];
    const unsigned char* mask = (const unsigned char*)d_in[4];

    float* out  = (float*)d_out;                       // [B,L,D] first
    float* attn = out + (size_t)B_ * L_ * D_;          // then [B,L,L]

    dim3 grid(B_ * (L_ / QT));   // 32 * 128 = 4096 blocks
    dim3 block(NT);              // 4 waves (wave32)
    sdpa_f32_wmma_kernel<<<grid, block, 0, stream>>>(q, k, v, diag, mask, out, attn);
}